// loss_t_41472204210650
// MI455X (gfx1250) — compile-verified
//
#include <hip/hip_runtime.h>

// ---------------------------------------------------------------------------
// Fused smoothness loss for MI455X (gfx1250).
//   x,t: f32[16,3,512,512].  Output: scalar f32.
//   Uses CDNA5 Tensor Data Mover (tensor_load_to_lds) to DMA 2D t-tiles into
//   LDS, overlapped with on-the-fly YCbCr (flat-reshape quirk) computation.
// ---------------------------------------------------------------------------

#define IMG     512
#define NB      16
#define TW      32
#define TH      32
#define HALO_W  36          // TW + 4 (dx in [-2,2])
#define HALO_H  34          // TH + 2 (dy in [0,2])
#define NPIX    (HALO_H * HALO_W)   // 1224
#define NBLOCKS ((IMG / TW) * (IMG / TH) * NB)  // 4096
#define NTOTAL  (NB * 3 * IMG * IMG)            // 12582912

// GAMMA * log2(e) = -(1/200) * 1.4426950408889634
#define KLOG2E  (-7.213475204444817e-3f)

typedef __attribute__((ext_vector_type(4))) unsigned int tdm_u32x4;
typedef __attribute__((ext_vector_type(8))) int          tdm_i32x8;
typedef __attribute__((ext_vector_type(4))) int          tdm_i32x4;

// Issue a 2D TDM load: global (strided rows) -> packed LDS tile.
__device__ __forceinline__ void tdm_load_tile_2d(unsigned lds_addr,
                                                 unsigned long long gaddr,
                                                 unsigned tensor_d0, unsigned tensor_d1,
                                                 unsigned tile_d0, unsigned tile_d1,
                                                 unsigned stride_elems)
{
    // ---- D# group 0 (128b): count=1, lds_addr, 57b global_addr, type=2 ----
    tdm_u32x4 g0;
    g0[0] = 1u;                                             // count=1 (valid), user mode
    g0[1] = lds_addr;                                       // LDS byte address
    g0[2] = (unsigned)(gaddr & 0xFFFFFFFFull);              // global_addr[31:0]
    g0[3] = ((unsigned)((gaddr >> 32) & 0x1FFFFFFull))      // global_addr[56:32]
            | (2u << 30);                                   // type = 2 ("image")
    // ---- D# group 1 (256b) ----
    tdm_i32x8 g1;
    g1[0] = (int)(2u << 16);                                // wg_mask=0, data_size=2 (4B)
    g1[1] = (int)((tensor_d0 & 0xFFFFu) << 16);             // tensor_dim0[15:0] @ b63:48
    g1[2] = (int)(((tensor_d0 >> 16) & 0xFFFFu)             // tensor_dim0[31:16]
                  | ((tensor_d1 & 0xFFFFu) << 16));         // tensor_dim1[15:0]
    g1[3] = (int)(((tensor_d1 >> 16) & 0xFFFFu)             // tensor_dim1[31:16]
                  | ((tile_d0 & 0xFFFFu) << 16));           // tile_dim0
    g1[4] = (int)(tile_d1 & 0xFFFFu);                       // tile_dim1; tile_dim2=0
    g1[5] = (int)stride_elems;                              // tensor_dim0_stride[31:0]
    g1[6] = 0;                                              // stride[47:32], dim1_stride lo
    g1[7] = 0;
    tdm_i32x4 gz = {0, 0, 0, 0};
#if defined(__clang_major__) && (__clang_major__ >= 23)
    tdm_i32x8 gz8 = {0, 0, 0, 0, 0, 0, 0, 0};
    __builtin_amdgcn_tensor_load_to_lds(g0, g1, gz, gz, gz8, 0);
#else
    __builtin_amdgcn_tensor_load_to_lds(g0, g1, gz, gz, 0);
#endif
}

// Per-offset scale: 2 / (B * (H-dy) * (W-|dx|))
__host__ __device__ constexpr float off_scale(int dy, int adx)
{
    return 2.0f / (16.0f * (float)(512 - dy) * (float)(512 - adx));
}

// One smoothness term; select on the FINAL value so garbage/NaN from
// guarded-off LDS reads can never reach the accumulator.
__device__ __forceinline__ float term_acc(const float4 ya, const float4 yb,
                                          const float4 ta, const float4 tb,
                                          bool ok, float sc, float acc)
{
    float d0 = ya.x - yb.x, d1 = ya.y - yb.y, d2 = ya.z - yb.z;
    float s2 = fmaf(d0, d0, fmaf(d1, d1, d2 * d2));
    float g  = fabsf(ta.x - tb.x) + fabsf(ta.y - tb.y) + fabsf(ta.z - tb.z);
    float w  = exp2f(s2 * KLOG2E);                  // v_exp_f32
    float tv = (w * g) * sc;
    return acc + (ok ? tv : 0.0f);
}

__global__ __launch_bounds__(256)
void loss_smooth_kernel(const float* __restrict__ x,
                        const float* __restrict__ t,
                        float* __restrict__ partial)
{
    __shared__ float  sstage[3][NPIX];    // TDM destination: t, channel-planar
    __shared__ float4 shY4[NPIX];         // y (ycbcr) packed per pixel
    __shared__ float4 shT4[NPIX];         // t packed per pixel
    __shared__ float  red[256];

    const int tid    = threadIdx.x;
    const int tileX0 = blockIdx.x * TW;
    const int gy0    = blockIdx.y * TH;
    const int b      = blockIdx.z;
    const int b3     = b * 3;
    const int gx0    = (tileX0 == 0) ? 0 : (tileX0 - 2);   // never read before buffer

    // ---- 1) wave 0 (provably uniform test -> scalar branch, so waves 1..7
    //         can never issue duplicate DMAs: TDM ignores EXEC) issues three
    //         TDM loads of the t channels (34x36 tiles, packed rows) --------
    const int waveid = __builtin_amdgcn_readfirstlane(tid) >> 5;
    if (waveid == 0) {
        const unsigned td0 = (unsigned)(IMG - gx0);   // right-edge OOB -> zero fill
        const unsigned td1 = (unsigned)(IMG - gy0);   // bottom-edge OOB -> zero fill
#pragma unroll
        for (int c = 0; c < 3; ++c) {
            unsigned lds = (unsigned)(unsigned long long)(&sstage[c][0]);
            unsigned long long ga = (unsigned long long)t +
                4ull * ((unsigned long long)((b3 + c) * IMG + gy0) * IMG + (unsigned)gx0);
            tdm_load_tile_2d(lds, ga, td0, td1, HALO_W, HALO_H, IMG);
        }
    }

    // ---- 2) overlap: compute y halo tile into LDS (flat-reshape quirk) ----
    for (int idx = tid; idx < NPIX; idx += 256) {
        int r   = idx / HALO_W;
        int cc  = idx - r * HALO_W;
        int gy  = gy0 + r;
        int gxc = gx0 + cc;
        float4 yv; yv.x = 0.f; yv.y = 0.f; yv.z = 0.f; yv.w = 0.f;
        if (gy < IMG && gxc < IMG) {
            int f0 = (b3 * IMG + gy) * IMG + gxc;
            float yc[3];
#pragma unroll
            for (int k = 0; k < 3; ++k) {
                unsigned f   = (unsigned)(f0 + k * (IMG * IMG));
                unsigned n   = f / 3u;
                unsigned rem = f - 3u * n;
                const float* xp = x + 3u * n;
                float x0 = xp[0], x1 = xp[1], x2 = xp[2];
                float c0 = (rem == 0u) ? 0.257f : ((rem == 1u) ? -0.148f : 0.439f);
                float c1 = (rem == 0u) ? 0.564f : ((rem == 1u) ? -0.291f : -0.368f);
                float c2 = (rem == 0u) ? 0.098f : ((rem == 1u) ? 0.439f : -0.071f);
                float cb = (rem == 0u) ? (16.0f / 255.0f) : (128.0f / 255.0f);
                yc[k] = fmaf(x0, c0, fmaf(x1, c1, fmaf(x2, c2, cb)));
            }
            yv.x = yc[0]; yv.y = yc[1]; yv.z = yc[2];
        }
        shY4[idx] = yv;
    }

    // Wave 0 waits for its DMAs; for all other waves TENSORcnt==0 -> no-op.
    __builtin_amdgcn_s_wait_tensorcnt(0);
    __syncthreads();

    // ---- 3) repack t: planar -> float4 (one ds_load_b128 per tap later) ----
    for (int idx = tid; idx < NPIX; idx += 256) {
        float4 tv;
        tv.x = sstage[0][idx]; tv.y = sstage[1][idx]; tv.z = sstage[2][idx]; tv.w = 0.f;
        shT4[idx] = tv;
    }
    __syncthreads();

    // ---- 4) smoothness terms: 4 horizontally adjacent pixels per thread ----
    const int ty  = tid >> 3;            // 0..31
    const int tx0 = (tid & 7) << 2;      // 0,4,...,28
    const int i   = gy0 + ty;            // global row of the 4 pixels
    const int jb  = tileX0 + tx0;        // global col of pixel p=0
    const int wcb = jb - 2 - gx0;        // LDS col of window slot 0

    float  acc = 0.f;
    float4 sy[4], st4[4];

    { // row dy = 0
        int base = ty * HALO_W + wcb;
        float4 wy[8], wt[8];
#pragma unroll
        for (int c = 0; c < 8; ++c) { wy[c] = shY4[base + c]; wt[c] = shT4[base + c]; }
#pragma unroll
        for (int p = 0; p < 4; ++p) { sy[p] = wy[p + 2]; st4[p] = wt[p + 2]; }
#pragma unroll
        for (int p = 0; p < 4; ++p) {
#pragma unroll
            for (int dx = 1; dx <= 2; ++dx) {
                bool ok = (unsigned)(jb + p + dx) < (unsigned)IMG;
                acc = term_acc(wy[p + 2 + dx], sy[p], wt[p + 2 + dx], st4[p],
                               ok, off_scale(0, dx), acc);
            }
        }
    }
#pragma unroll
    for (int r = 1; r <= 2; ++r) { // rows dy = 1, 2
        int base = (ty + r) * HALO_W + wcb;
        float4 wy[8], wt[8];
#pragma unroll
        for (int c = 0; c < 8; ++c) { wy[c] = shY4[base + c]; wt[c] = shT4[base + c]; }
        bool rok = (i + r) < IMG;
#pragma unroll
        for (int p = 0; p < 4; ++p) {
#pragma unroll
            for (int dxi = 0; dxi < 5; ++dxi) {
                const int dx = dxi - 2;
                bool ok = rok && ((unsigned)(jb + p + dx) < (unsigned)IMG);
                acc = term_acc(wy[p + 2 + dx], sy[p], wt[p + 2 + dx], st4[p],
                               ok, off_scale(r, dx < 0 ? -dx : dx), acc);
            }
        }
    }

    // ---- 5) fidelity term: 1.5 * mean((x-t)^2), core pixels only ----
    float accF = 0.f;
#pragma unroll
    for (int k = 0; k < 3; ++k) {
        const float4 xv = *reinterpret_cast<const float4*>(
            x + (((size_t)(b3 + k) * IMG + i) * IMG + jb));
        float t0 = (k == 0) ? st4[0].x : (k == 1) ? st4[0].y : st4[0].z;
        float t1 = (k == 0) ? st4[1].x : (k == 1) ? st4[1].y : st4[1].z;
        float t2 = (k == 0) ? st4[2].x : (k == 1) ? st4[2].y : st4[2].z;
        float t3 = (k == 0) ? st4[3].x : (k == 1) ? st4[3].y : st4[3].z;
        float e;
        e = xv.x - t0; accF = fmaf(e, e, accF);
        e = xv.y - t1; accF = fmaf(e, e, accF);
        e = xv.z - t2; accF = fmaf(e, e, accF);
        e = xv.w - t3; accF = fmaf(e, e, accF);
    }
    float total = fmaf(accF, 1.5f / (float)NTOTAL, acc);

    // ---- 6) deterministic block reduction -> one partial per block ----
    red[tid] = total;
    __syncthreads();
#pragma unroll
    for (int o = 128; o > 0; o >>= 1) {
        if (tid < o) red[tid] += red[tid + o];
        __syncthreads();
    }
    if (tid == 0) {
        int bid = (blockIdx.z * gridDim.y + blockIdx.y) * gridDim.x + blockIdx.x;
        partial[bid] = red[0];
    }
}

__global__ __launch_bounds__(256)
void loss_finalize_kernel(const float* __restrict__ ws, float* __restrict__ out)
{
    __shared__ float red[256];
    float s = 0.f;
    for (int idx = threadIdx.x; idx < NBLOCKS; idx += 256) s += ws[idx];
    red[threadIdx.x] = s;
    __syncthreads();
#pragma unroll
    for (int o = 128; o > 0; o >>= 1) {
        if ((int)threadIdx.x < o) red[threadIdx.x] += red[threadIdx.x + o];
        __syncthreads();
    }
    if (threadIdx.x == 0) out[0] = red[0];
}

extern "C" void kernel_launch(void* const* d_in, const int* in_sizes, int n_in,
                              void* d_out, int out_size, void* d_ws, size_t ws_size,
                              hipStream_t stream)
{
    (void)in_sizes; (void)n_in; (void)out_size; (void)ws_size;
    const float* x = (const float*)d_in[0];
    const float* t = (const float*)d_in[1];
    float* out = (float*)d_out;
    float* ws  = (float*)d_ws;                  // 4096 floats of scratch

    dim3 grid(IMG / TW, IMG / TH, NB);          // (16,16,16) = 4096 blocks
    loss_smooth_kernel<<<grid, 256, 0, stream>>>(x, t, ws);
    loss_finalize_kernel<<<1, 256, 0, stream>>>(ws, out);
}